// SlidingWindowAttention_6279242187106
// MI455X (gfx1250) — compile-verified
//
#include <hip/hip_runtime.h>

// ---------------------------------------------------------------------------
// Sliding-window attention, fp32, MI455X (gfx1250, wave32, WMMA).
//   B=2, S=4096, D=128, window_size=128 (half=64 -> 129 taps, union=144/tile)
// Kernel 1: q/k/v = x @ W + b           via V_WMMA_F32_16X16X4_F32
// Kernel 2: windowed scores -> softmax -> @V, fp32 WMMA, LDS staged with
//           GLOBAL_LOAD_ASYNC_TO_LDS_B128 (ASYNCcnt) on the k/v/q tiles.
// ---------------------------------------------------------------------------

typedef __attribute__((ext_vector_type(2))) float v2f;
typedef __attribute__((ext_vector_type(8))) float v8f;

#define BATCH 2
#define SEQ   4096
#define DIM   128
#define HALF  64
#define UNION 144           // 16 + 2*HALF
#define SCW   148           // padded score-row stride (floats, 8B aligned)
#define NEG_INF_F (-1e30f)

// LDS byte offsets, kernel 2 (single dynamic allocation -> base offset 0):
#define QS_OFF 0u                                  // 16   rows * 512 B
#define KS_OFF 8192u                               // 144  rows * 512 B
#define VS_OFF (8192u + 73728u)                    // 144  rows * 512 B

static __device__ __forceinline__ v8f wmma_f32(v2f a, v2f b, v8f c) {
  // D = A(16x4, f32) * B(4x16, f32) + C(16x16, f32)
  return __builtin_amdgcn_wmma_f32_16x16x4_f32(false, a, false, b,
                                               (short)0, c, false, false);
}

// One wave32 instruction moves 32 lanes x 16 B = 512 B global -> LDS,
// tracked by ASYNCcnt, no VGPR round-trip.
static __device__ __forceinline__ void async_row_b128(unsigned lds_byte_off,
                                                      const void* gaddr) {
  asm volatile("global_load_async_to_lds_b128 %0, %1, off"
               :: "v"(lds_byte_off), "v"(gaddr)
               : "memory");
}
static __device__ __forceinline__ void wait_async0() {
  asm volatile("s_wait_asynccnt 0x0" ::: "memory");
}

// ---------------------------------------------------------------------------
// Kernel 1: fused QKV projection. grid = (B*S)/16 blocks, 256 threads (8 waves)
// Each wave owns one 16-wide slice of the D=128 output columns.
// ---------------------------------------------------------------------------
__global__ void swa_qkv_kernel(const float* __restrict__ x,
                               const float* __restrict__ Wq, const float* __restrict__ bq,
                               const float* __restrict__ Wk, const float* __restrict__ bk,
                               const float* __restrict__ Wv, const float* __restrict__ bv,
                               float* __restrict__ qo, float* __restrict__ ko,
                               float* __restrict__ vo) {
  __shared__ float xs[16 * DIM];                 // 8 KB x tile (LDS offset 0)
  const int row0 = blockIdx.x * 16;              // flat row in [0, B*S)
  const int tid  = threadIdx.x;
  const int wave = tid >> 5;                     // 0..7 -> N-tile
  const int lane = tid & 31;

  // Pull weights toward L2 early (global_prefetch_b8).
  __builtin_prefetch(&Wq[tid * 64], 0, 1);
  __builtin_prefetch(&Wk[tid * 64], 0, 1);
  __builtin_prefetch(&Wv[tid * 64], 0, 1);

  // Async-stage the 16x128 x tile: one 512 B row per wave-instruction.
  for (int r = wave; r < 16; r += 8)
    async_row_b128((unsigned)(r * 512 + lane * 16),
                   x + (size_t)(row0 + r) * DIM + lane * 4);
  wait_async0();
  __syncthreads();

  const int half = lane >> 4;                    // A/B lane halves (K pairs)
  const int l15  = lane & 15;
  const int n0   = wave * 16;

  const float* Ws[3]   = {Wq, Wk, Wv};
  const float* bias[3] = {bq, bk, bv};
  float*       outs[3] = {qo, ko, vo};

#pragma unroll
  for (int m = 0; m < 3; ++m) {
    const float* W = Ws[m];
    v8f acc = {};
#pragma unroll 4
    for (int kk = 0; kk < DIM / 4; ++kk) {
      const int cb = kk * 4 + half * 2;          // K column base for this lane
      v2f a = *(const v2f*)&xs[l15 * DIM + cb];  // A: x[M=l15][K=cb..cb+1]
      v2f b;                                     // B: W[K][N=n0+l15]
      b.x = W[(cb + 0) * DIM + n0 + l15];
      b.y = W[(cb + 1) * DIM + n0 + l15];
      acc = wmma_f32(a, b, acc);
    }
    const float bv_ = bias[m][n0 + l15];
    float* o = outs[m];
#pragma unroll
    for (int r = 0; r < 8; ++r) {
      const int M = r + half * 8;                // C/D layout: VGPR r -> M=r / r+8
      o[(row0 + M) * DIM + n0 + l15] = acc[r] + bv_;
    }
  }
}

// ---------------------------------------------------------------------------
// Kernel 2: sliding-window attention over one 16-query tile per block.
// grid = B*(S/16) blocks, 256 threads (8 waves).
// LDS: q-tile 8KB | k-window 72KB | v-window 72KB | scores 9.3KB  (~161 KB,
// sized for the 320 KB/WGP CDNA5 LDS). k/v/q staged via async-to-LDS.
// ---------------------------------------------------------------------------
__global__ void swa_attn_kernel(const float* __restrict__ q,
                                const float* __restrict__ k,
                                const float* __restrict__ v,
                                float* __restrict__ out) {
  extern __shared__ float lds[];
  float* qs = lds;                               // 16   * 128   (byte off 0)
  float* ks = qs + 16 * DIM;                     // 144  * 128   (byte off 8192)
  float* vs = ks + UNION * DIM;                  // 144  * 128   (byte off 81920)
  float* sc = vs + UNION * DIM;                  // 16   * SCW

  const int b    = blockIdx.x / (SEQ / 16);
  const int t0   = (blockIdx.x % (SEQ / 16)) * 16;   // query tile base (in batch)
  const int base = b * SEQ;                          // flat row base of batch
  const int tid  = threadIdx.x;
  const int wave = tid >> 5;
  const int lane = tid & 31;

  // ---- async-stage q tile + 144-key union window of k/v ------------------
  for (int r = wave; r < 16; r += 8)             // q tile: 16 rows
    async_row_b128(QS_OFF + (unsigned)(r * 512 + lane * 16),
                   q + (size_t)(base + t0 + r) * DIM + lane * 4);
  for (int u = wave; u < UNION; u += 8) {        // k/v window: 144 rows each
    const int j = t0 - HALF + u;                 // wave-uniform validity
    const unsigned rofs = (unsigned)(u * 512 + lane * 16);
    if (j >= 0 && j < SEQ) {
      async_row_b128(KS_OFF + rofs, k + (size_t)(base + j) * DIM + lane * 4);
      async_row_b128(VS_OFF + rofs, v + (size_t)(base + j) * DIM + lane * 4);
    } else {                                     // OOB rows: zero-fill
      const float4 z = {0.0f, 0.0f, 0.0f, 0.0f};
      *(float4*)&ks[u * DIM + lane * 4] = z;
      *(float4*)&vs[u * DIM + lane * 4] = z;
    }
  }
  wait_async0();
  __syncthreads();

  const int half = lane >> 4;
  const int l15  = lane & 15;
  const float scale = 0.08838834764831845f;      // 1/sqrt(128)

  // ---- scores: S[M][u] = q[M,:] . k[u,:]  (9 key tiles over 8 waves) ------
  for (int n = wave; n < UNION / 16; n += 8) {   // wave-uniform loop (EXEC=all1)
    const int n0 = n * 16;
    v8f acc = {};
#pragma unroll 4
    for (int kk = 0; kk < DIM / 4; ++kk) {
      const int cb = kk * 4 + half * 2;
      v2f a  = *(const v2f*)&qs[l15 * DIM + cb];         // q[M][K pair]
      v2f bm = *(const v2f*)&ks[(n0 + l15) * DIM + cb];  // k^T[K pair][N=key]
      acc = wmma_f32(a, bm, acc);
    }
    const int u = n0 + l15;
    const int j = t0 - HALF + u;
#pragma unroll
    for (int r = 0; r < 8; ++r) {
      const int M = r + half * 8;
      const bool valid = (u >= M) && (u <= M + 2 * HALF) && (j >= 0) && (j < SEQ);
      sc[M * SCW + u] = valid ? acc[r] * scale : NEG_INF_F;
    }
  }
  __syncthreads();

  // ---- row softmax (16 rows, one thread each; no WMMA in divergent code) --
  if (tid < 16) {
    float mx = NEG_INF_F;
    for (int u = 0; u < UNION; ++u) mx = fmaxf(mx, sc[tid * SCW + u]);
    float sum = 0.0f;
    for (int u = 0; u < UNION; ++u) {
      const float e = __expf(sc[tid * SCW + u] - mx);
      sc[tid * SCW + u] = e;
      sum += e;
    }
    const float inv = 1.0f / sum;
    for (int u = 0; u < UNION; ++u) sc[tid * SCW + u] *= inv;
  }
  __syncthreads();

  // ---- out = P(16x144) @ Vwin(144x128)  (one 16-wide N-tile per wave) -----
  for (int n = wave; n < DIM / 16; n += 8) {
    const int n0 = n * 16;
    v8f acc = {};
#pragma unroll 4
    for (int kk = 0; kk < UNION / 4; ++kk) {
      const int cb = kk * 4 + half * 2;
      v2f a = *(const v2f*)&sc[l15 * SCW + cb];  // P[M][K pair]
      v2f bm;                                    // V[K][N=n0+l15]
      bm.x = vs[(cb + 0) * DIM + n0 + l15];
      bm.y = vs[(cb + 1) * DIM + n0 + l15];
      acc = wmma_f32(a, bm, acc);
    }
#pragma unroll
    for (int r = 0; r < 8; ++r) {
      const int M = r + half * 8;
      out[(base + t0 + M) * DIM + n0 + l15] = acc[r];
    }
  }
}

// ---------------------------------------------------------------------------
extern "C" void kernel_launch(void* const* d_in, const int* in_sizes, int n_in,
                              void* d_out, int out_size, void* d_ws, size_t ws_size,
                              hipStream_t stream) {
  (void)in_sizes; (void)n_in; (void)out_size; (void)ws_size;
  const float* x  = (const float*)d_in[0];
  const float* Wq = (const float*)d_in[1];
  const float* bq = (const float*)d_in[2];
  const float* Wk = (const float*)d_in[3];
  const float* bk = (const float*)d_in[4];
  const float* Wv = (const float*)d_in[5];
  const float* bv = (const float*)d_in[6];
  // d_in[7] = window_size (128), baked into the kernel constants.
  float* out = (float*)d_out;

  const size_t n = (size_t)BATCH * SEQ * DIM;    // elements per q/k/v buffer
  float* qbuf = (float*)d_ws;
  float* kbuf = qbuf + n;
  float* vbuf = kbuf + n;

  const int tiles = (BATCH * SEQ) / 16;          // 512

  swa_qkv_kernel<<<tiles, 256, 0, stream>>>(x, Wq, bq, Wk, bk, Wv, bv,
                                            qbuf, kbuf, vbuf);

  const size_t ldsBytes =
      (16 * DIM + 2 * UNION * DIM + 16 * SCW) * sizeof(float); // ~161.25 KB
  swa_attn_kernel<<<tiles, 256, ldsBytes, stream>>>(qbuf, kbuf, vbuf, out);
}